// GMConv_14714557956308
// MI455X (gfx1250) — compile-verified
//
#include <hip/hip_runtime.h>
#include <hip/hip_bf16.h>
#include <stdint.h>

// GM convolution + moment-matching merge, algebraically collapsed:
//   S  = sum_i w_i
//   mb = sum_i w_i m_i / S
//   Cb = sum_i w_i (C_i + m_i m_i^T)/S - mb mb^T
//   W[o,k]   = k_w[o,k] * S
//   mu[o,k]  = anchor[o] + k_means[o,k] + mb
//   cov[o,k] = (R diag(s)) (R diag(s))^T + Cb
// anchor[o] = means[idx of o-th largest weight] (stable top-k via rank count).
// Launch-latency bound; no GEMM structure -> WMMA intentionally not used.
// CDNA5 paths used: global_load_async_to_lds_b128 + s_wait_asynccnt,
// global_prefetch_b8 (via __builtin_prefetch), wave32 shuffle reductions.

#define No_ 256
#define Nk_ 32
#define Ni_ 512
#define OPB 8          // output rows (o) per block
#define BLK 256        // threads per block = 8 wave32
#define NWAVE (BLK / 32)

__global__ __launch_bounds__(BLK) void GMConv_gfx1250_kernel(
    const float* __restrict__ weights,    // (Ni)
    const float* __restrict__ means,      // (Ni,3)
    const float* __restrict__ covars,     // (Ni,3,3)
    const float* __restrict__ k_weights,  // (No,Nk)
    const float* __restrict__ k_means,    // (No,Nk,3)
    const float* __restrict__ k_scales,   // (No,Nk,3)
    const float* __restrict__ k_quats,    // (No,Nk,4)
    float* __restrict__ outW,             // (No,Nk)
    float* __restrict__ outMu,            // (No,Nk,3)
    float* __restrict__ outCov)           // (No,Nk,3,3)
{
    __shared__ __align__(16) float sW[Ni_];        // staged weights (async DMA target)
    __shared__ __align__(16) float sRed[NWAVE][16];// per-wave partials (13 used, padded)
    __shared__ float sAnchor[OPB][3];              // anchors for this block's o-range

    const int tid = threadIdx.x;
    const int o0  = blockIdx.x * OPB;
    const int lo  = tid >> 5;             // 0..7  : local o
    const int kk  = tid & 31;             // 0..31 : k
    const int g   = (o0 + lo) * Nk_ + kk; // flat (o,k)

    // ---- Prefetch this thread's phase-B operands (hide HBM latency behind
    //      the rank loop). Lowers to global_prefetch_b8 on gfx1250.
    __builtin_prefetch(&k_quats[4 * g], 0, 3);
    __builtin_prefetch(&k_means[3 * g], 0, 3);
    __builtin_prefetch(&k_scales[3 * g], 0, 3);
    __builtin_prefetch(&k_weights[g], 0, 3);

    // ---- Stage weights[] (2 KB) into LDS with CDNA5 async global->LDS DMA:
    //      128 lanes x b128 = 512 dwords.
    if (tid < Ni_ / 4) {
        unsigned lds = (unsigned)(uintptr_t)(&sW[tid * 4]);
        unsigned off = (unsigned)(tid * 16);
        asm volatile("global_load_async_to_lds_b128 %0, %1, %2"
                     :: "v"(lds), "v"(off), "s"(weights) : "memory");
    }
    asm volatile("s_wait_asynccnt 0x0" ::: "memory");
    __syncthreads();   // barrier #1: sW visible to all waves

    // ---- Per-thread partial sums over its 2 input components:
    //      acc = [S, S*m(3), sum w*(C + m m^T)(9)]
    float acc[13];
    #pragma unroll
    for (int v = 0; v < 13; ++v) acc[v] = 0.0f;

    #pragma unroll
    for (int e = 0; e < 2; ++e) {
        const int i = tid + e * BLK;
        const float w  = sW[i];
        const float m0 = means[3 * i + 0];
        const float m1 = means[3 * i + 1];
        const float m2 = means[3 * i + 2];
        acc[0] += w;
        acc[1] += w * m0;  acc[2] += w * m1;  acc[3] += w * m2;
        const float mm[3] = {m0, m1, m2};
        #pragma unroll
        for (int r = 0; r < 3; ++r)
            #pragma unroll
            for (int c = 0; c < 3; ++c)
                acc[4 + 3 * r + c] += w * (covars[9 * i + 3 * r + c] + mm[r] * mm[c]);
    }

    // ---- wave32 shuffle reduction (deterministic fixed order), then one
    //      13-float LDS store per wave.
    #pragma unroll
    for (int v = 0; v < 13; ++v) {
        float x = acc[v];
        x += __shfl_xor(x, 16, 32);
        x += __shfl_xor(x,  8, 32);
        x += __shfl_xor(x,  4, 32);
        x += __shfl_xor(x,  2, 32);
        x += __shfl_xor(x,  1, 32);
        acc[v] = x;
    }
    if (kk == 0) {
        #pragma unroll
        for (int v = 0; v < 13; ++v) sRed[lo][v] = acc[v];
    }

    // ---- Stable top-k ranks (only the 8 ranks this block needs matter).
    //      rank(i) = #{j : w_j > w_i  or (w_j == w_i and j < i)}
    //      Matches jax.lax.top_k stable-descending order.
    #pragma unroll
    for (int e = 0; e < 2; ++e) {
        const int i = tid + e * BLK;
        const float wi = sW[i];
        int r = 0;
        for (int j = 0; j < Ni_; ++j) {
            const float wj = sW[j];
            r += ((wj > wi) || (wj == wi && j < i)) ? 1 : 0;
        }
        if (r >= o0 && r < o0 + OPB) {      // ranks are unique -> no conflicts
            sAnchor[r - o0][0] = means[3 * i + 0];
            sAnchor[r - o0][1] = means[3 * i + 1];
            sAnchor[r - o0][2] = means[3 * i + 2];
        }
    }
    __syncthreads();   // barrier #2: sRed + sAnchor visible

    // ---- Combine the 8 per-wave partials (broadcast LDS reads, fixed order).
    float st[13];
    #pragma unroll
    for (int v = 0; v < 13; ++v) {
        float t = sRed[0][v];
        #pragma unroll
        for (int w = 1; w < NWAVE; ++w) t += sRed[w][v];
        st[v] = t;
    }

    // ---- Phase B: one (o,k) pair per thread.
    const float S    = st[0];
    const float invS = 1.0f / S;
    const float mb0  = st[1] * invS;
    const float mb1  = st[2] * invS;
    const float mb2  = st[3] * invS;
    const float mb[3] = {mb0, mb1, mb2};
    float Cb[9];
    #pragma unroll
    for (int r = 0; r < 3; ++r)
        #pragma unroll
        for (int c = 0; c < 3; ++c)
            Cb[3 * r + c] = st[4 + 3 * r + c] * invS - mb[r] * mb[c];

    const float kw  = k_weights[g];
    const float km0 = k_means[3 * g + 0];
    const float km1 = k_means[3 * g + 1];
    const float km2 = k_means[3 * g + 2];
    const float s0  = k_scales[3 * g + 0];
    const float s1  = k_scales[3 * g + 1];
    const float s2  = k_scales[3 * g + 2];
    const float q0  = k_quats[4 * g + 0];
    const float q1  = k_quats[4 * g + 1];
    const float q2  = k_quats[4 * g + 2];
    const float q3  = k_quats[4 * g + 3];

    const float qn = 1.0f / sqrtf(q0 * q0 + q1 * q1 + q2 * q2 + q3 * q3);
    const float qw = q0 * qn, qx = q1 * qn, qy = q2 * qn, qz = q3 * qn;

    const float r00 = 1.0f - 2.0f * (qy * qy + qz * qz);
    const float r01 = 2.0f * (qx * qy - qw * qz);
    const float r02 = 2.0f * (qx * qz + qw * qy);
    const float r10 = 2.0f * (qx * qy + qw * qz);
    const float r11 = 1.0f - 2.0f * (qx * qx + qz * qz);
    const float r12 = 2.0f * (qy * qz - qw * qx);
    const float r20 = 2.0f * (qx * qz - qw * qy);
    const float r21 = 2.0f * (qy * qz + qw * qx);
    const float r22 = 1.0f - 2.0f * (qx * qx + qy * qy);

    const float m00 = r00 * s0, m01 = r01 * s1, m02 = r02 * s2;
    const float m10 = r10 * s0, m11 = r11 * s1, m12 = r12 * s2;
    const float m20 = r20 * s0, m21 = r21 * s1, m22 = r22 * s2;

    const float P00 = m00 * m00 + m01 * m01 + m02 * m02;
    const float P01 = m00 * m10 + m01 * m11 + m02 * m12;
    const float P02 = m00 * m20 + m01 * m21 + m02 * m22;
    const float P11 = m10 * m10 + m11 * m11 + m12 * m12;
    const float P12 = m10 * m20 + m11 * m21 + m12 * m22;
    const float P22 = m20 * m20 + m21 * m21 + m22 * m22;

    outW[g] = kw * S;

    outMu[3 * g + 0] = sAnchor[lo][0] + km0 + mb0;
    outMu[3 * g + 1] = sAnchor[lo][1] + km1 + mb1;
    outMu[3 * g + 2] = sAnchor[lo][2] + km2 + mb2;

    float* cv = &outCov[9 * g];
    cv[0] = P00 + Cb[0];  cv[1] = P01 + Cb[1];  cv[2] = P02 + Cb[2];
    cv[3] = P01 + Cb[3];  cv[4] = P11 + Cb[4];  cv[5] = P12 + Cb[5];
    cv[6] = P02 + Cb[6];  cv[7] = P12 + Cb[7];  cv[8] = P22 + Cb[8];
}

extern "C" void kernel_launch(void* const* d_in, const int* in_sizes, int n_in,
                              void* d_out, int out_size, void* d_ws, size_t ws_size,
                              hipStream_t stream) {
    // setup_inputs() order: weights, means, covars, features(unused),
    //                       k_weights, k_means, k_scales, k_quats  (all f32)
    const float* weights   = (const float*)d_in[0];
    const float* means     = (const float*)d_in[1];
    const float* covars    = (const float*)d_in[2];
    const float* k_weights = (const float*)d_in[4];
    const float* k_means   = (const float*)d_in[5];
    const float* k_scales  = (const float*)d_in[6];
    const float* k_quats   = (const float*)d_in[7];

    float* outW   = (float*)d_out;                       // 8192
    float* outMu  = outW  + No_ * Nk_;                   // +24576
    float* outCov = outMu + No_ * Nk_ * 3;               // +73728

    dim3 grid(No_ / OPB);   // 32 blocks
    dim3 block(BLK);        // 256 threads = 8 wave32
    GMConv_gfx1250_kernel<<<grid, block, 0, stream>>>(
        weights, means, covars, k_weights, k_means, k_scales, k_quats,
        outW, outMu, outCov);
}